// SinkhornDistance_738734375240
// MI455X (gfx1250) — compile-verified
//
#include <hip/hip_runtime.h>
#include <math.h>

// ---------------- problem constants ----------------
#define NB   8
#define P    2048
#define D    64
#define EPSF 0.1f
#define INV_EPS 10.0f
#define THRESH 0.1f
#define MAX_ITER 50
// log(1/2048 + 1e-8)
#define LOG_MU (-7.6245985f)
#define LOG_NU (-7.6245985f)
#define RCHUNKS 16

typedef __attribute__((ext_vector_type(2))) float v2f;
typedef __attribute__((ext_vector_type(8))) float v8f;

// ---------------- init: zero u, v, done ----------------
__global__ void sk_init_kernel(float* u, float* v, int* done) {
    int gid = blockIdx.x * blockDim.x + threadIdx.x;
    if (gid < NB * P) { u[gid] = 0.0f; v[gid] = 0.0f; }
    if (gid == 0) *done = 0;
}

// ---------------- row norms of x and y ----------------
__global__ void sk_norms_kernel(const float* __restrict__ x, const float* __restrict__ y,
                                float* __restrict__ xn, float* __restrict__ yn) {
    int gid = blockIdx.x * blockDim.x + threadIdx.x;   // [0, 2*NB*P)
    const int total = NB * P;
    const float* src = (gid < total) ? x : y;
    float*       dst = (gid < total) ? xn : yn;
    int idx = (gid < total) ? gid : gid - total;
    const float4* p = (const float4*)(src + (size_t)idx * D);
    float s = 0.0f;
#pragma unroll
    for (int k = 0; k < D / 4; ++k) {
        float4 q = p[k];
        s += q.x * q.x + q.y * q.y + q.z * q.z + q.w * q.w;
    }
    dst[idx] = s;
}

// ---------------- C = xn_i + yn_j - 2 * (x . y^T)  via V_WMMA_F32_16X16X4_F32 ----------------
// one wave -> one 16x16 tile of C; K=64 -> 16 chained WMMAs
__global__ void sk_cost_wmma_kernel(const float* __restrict__ x, const float* __restrict__ y,
                                    const float* __restrict__ xn, const float* __restrict__ yn,
                                    float* __restrict__ C) {
    int wave = (int)((blockIdx.x * blockDim.x + threadIdx.x) >> 5);
    int lane = threadIdx.x & 31;
    // 128*128 = 16384 tiles per batch
    int n  = wave >> 14;
    int r  = wave & 16383;
    int tm = r >> 7;
    int tn = r & 127;

    const float* xb = x + (size_t)n * P * D;
    const float* yb = y + (size_t)n * P * D;

    int half = lane >> 4;          // 0: K 0..1, 1: K 2..3
    int l15  = lane & 15;
    int arow = tm * 16 + l15;      // M index this lane loads
    int brow = tn * 16 + l15;      // N index (row of y) this lane loads
    int kofs = half * 2;

    v8f acc = {};
#pragma unroll
    for (int kk = 0; kk < D; kk += 4) {
        v2f a = *(const v2f*)(xb + (size_t)arow * D + kk + kofs);
        v2f b = *(const v2f*)(yb + (size_t)brow * D + kk + kofs);
        acc = __builtin_amdgcn_wmma_f32_16x16x4_f32(false, a, false, b,
                                                    (short)0, acc, false, false);
    }

    // epilogue: C[m][col] = xn[m] + yn[col] - 2*acc
    float cn = yn[(size_t)n * P + tn * 16 + l15];      // column norm (per-lane)
    const float* xnb = xn + (size_t)n * P;
    float* Cb = C + (size_t)n * P * P;
#pragma unroll
    for (int p8 = 0; p8 < 8; ++p8) {
        int m = tm * 16 + p8 + 8 * half;
        float val = xnb[m] + cn - 2.0f * acc[p8];
        Cb[(size_t)m * P + tn * 16 + l15] = val;
    }
}

// online logsumexp: absorb one value (1 exp per element, minimal)
__device__ __forceinline__ void ms_point(float& m, float& s, float a) {
    if (a > m) { s = s * __expf(m - a) + 1.0f; m = a; }
    else       { s += __expf(a - m); }
}
// merge two (m,s) states
__device__ __forceinline__ void ms_merge(float& m, float& s, float om, float os) {
    if (om > m) { s = s * __expf(m - om) + os; m = om; }
    else        { s = s + os * __expf(om - m); }
}

// ---------------- u update: one wave per row, float4 loads, 4-way ILP states ----------------
__global__ void sk_update_u_kernel(const float* __restrict__ C, float* __restrict__ u,
                                   const float* __restrict__ v, float* __restrict__ du,
                                   const int* __restrict__ done) {
    if (*done) return;
    int wave = (int)((blockIdx.x * blockDim.x + threadIdx.x) >> 5);
    int lane = threadIdx.x & 31;
    int n = wave >> 11;
    int i = wave & (P - 1);
    const float4* Crow4 = (const float4*)(C + ((size_t)n * P + i) * P);
    const float4* vb4   = (const float4*)(v + (size_t)n * P);

    float m0 = -3.0e38f, m1 = -3.0e38f, m2 = -3.0e38f, m3 = -3.0e38f;
    float s0 = 0.0f, s1 = 0.0f, s2 = 0.0f, s3 = 0.0f;
#pragma unroll 4
    for (int t = lane; t < P / 4; t += 32) {      // 16 iterations/lane
        float4 c4 = Crow4[t];
        float4 v4 = vb4[t];
        ms_point(m0, s0, (v4.x - c4.x) * INV_EPS);
        ms_point(m1, s1, (v4.y - c4.y) * INV_EPS);
        ms_point(m2, s2, (v4.z - c4.z) * INV_EPS);
        ms_point(m3, s3, (v4.w - c4.w) * INV_EPS);
    }
    ms_merge(m0, s0, m1, s1);
    ms_merge(m2, s2, m3, s3);
    ms_merge(m0, s0, m2, s2);
#pragma unroll
    for (int off = 16; off; off >>= 1) {
        float om = __shfl_xor(m0, off);
        float os = __shfl_xor(s0, off);
        ms_merge(m0, s0, om, os);
    }
    if (lane == 0) {
        float lse = m0 + __logf(s0);
        float uo  = u[(size_t)n * P + i];
        float un  = EPSF * (LOG_MU - lse);
        u[(size_t)n * P + i]  = un;
        du[(size_t)n * P + i] = fabsf(un - uo);
    }
}

// ---------------- v update stage 1: partial column LSE; thread owns 4 columns ------------
// grid = 16 column-blocks (1024 cols each) x RCHUNKS row-chunks
__global__ void sk_v_partial_kernel(const float* __restrict__ C, const float* __restrict__ u,
                                    float* __restrict__ pm, float* __restrict__ ps,
                                    const int* __restrict__ done) {
    if (*done) return;
    int cb     = blockIdx.x & 15;
    int rchunk = blockIdx.x >> 4;
    int gcol0 = cb * 1024 + threadIdx.x * 4;      // first of 4 consecutive global cols
    int n = gcol0 >> 11;
    int j = gcol0 & (P - 1);
    const float* Cb = C + (size_t)n * P * P;
    const float* ub = u + (size_t)n * P;

    float m0 = -3.0e38f, m1 = -3.0e38f, m2 = -3.0e38f, m3 = -3.0e38f;
    float s0 = 0.0f, s1 = 0.0f, s2 = 0.0f, s3 = 0.0f;
    int i0 = rchunk * (P / RCHUNKS);
#pragma unroll 4
    for (int i = i0; i < i0 + P / RCHUNKS; ++i) {   // 128 rows/chunk
        float4 c4 = *(const float4*)(Cb + (size_t)i * P + j);
        float ui = ub[i] * INV_EPS;                 // wave-uniform scalar load
        ms_point(m0, s0, ui - c4.x * INV_EPS);
        ms_point(m1, s1, ui - c4.y * INV_EPS);
        ms_point(m2, s2, ui - c4.z * INV_EPS);
        ms_point(m3, s3, ui - c4.w * INV_EPS);
    }
    size_t base = (size_t)rchunk * (NB * P) + gcol0;
    *(float4*)(pm + base) = make_float4(m0, m1, m2, m3);
    *(float4*)(ps + base) = make_float4(s0, s1, s2, s3);
}

// ---------------- v update stage 2: combine chunks (fixed order, deterministic) ----------
__global__ void sk_v_combine_kernel(float* __restrict__ v, const float* __restrict__ pm,
                                    const float* __restrict__ ps, const int* __restrict__ done) {
    if (*done) return;
    int gcol = blockIdx.x * blockDim.x + threadIdx.x;   // [0, NB*P)
    float m = -3.0e38f, s = 0.0f;
#pragma unroll
    for (int r = 0; r < RCHUNKS; ++r)
        ms_merge(m, s, pm[(size_t)r * (NB * P) + gcol], ps[(size_t)r * (NB * P) + gcol]);
    v[gcol] = EPSF * (LOG_NU - (m + __logf(s)));
}

// ---------------- convergence check (single block, fixed-order reduction) ----------------
__global__ void sk_check_kernel(const float* __restrict__ du, int* done) {
    if (*done) return;
    __shared__ float red[256];
    float s = 0.0f;
    for (int k = threadIdx.x; k < NB * P; k += 256) s += du[k];
    red[threadIdx.x] = s;
    __syncthreads();
    for (int w = 128; w; w >>= 1) {
        if ((int)threadIdx.x < w) red[threadIdx.x] += red[threadIdx.x + w];
        __syncthreads();
    }
    if (threadIdx.x == 0) {
        if (red[0] / (float)NB < THRESH) *done = 1;
    }
}

// ---------------- final cost: per-row partial sums (one wave per row, float4) -----------
__global__ void sk_cost_partial_kernel(const float* __restrict__ C, const float* __restrict__ u,
                                       const float* __restrict__ v, float* __restrict__ rowsum) {
    int wave = (int)((blockIdx.x * blockDim.x + threadIdx.x) >> 5);
    int lane = threadIdx.x & 31;
    int n = wave >> 11;
    int i = wave & (P - 1);
    const float4* Crow4 = (const float4*)(C + ((size_t)n * P + i) * P);
    const float4* vb4   = (const float4*)(v + (size_t)n * P);
    float ui = u[(size_t)n * P + i] * INV_EPS;
    float s0 = 0.0f, s1 = 0.0f, s2 = 0.0f, s3 = 0.0f;
#pragma unroll 4
    for (int t = lane; t < P / 4; t += 32) {
        float4 c4 = Crow4[t];
        float4 v4 = vb4[t];
        s0 += __expf(ui + (v4.x - c4.x) * INV_EPS) * c4.x;
        s1 += __expf(ui + (v4.y - c4.y) * INV_EPS) * c4.y;
        s2 += __expf(ui + (v4.z - c4.z) * INV_EPS) * c4.z;
        s3 += __expf(ui + (v4.w - c4.w) * INV_EPS) * c4.w;
    }
    float s = (s0 + s1) + (s2 + s3);
#pragma unroll
    for (int off = 16; off; off >>= 1) s += __shfl_xor(s, off);
    if (lane == 0) rowsum[(size_t)n * P + i] = s;
}

// ---------------- final cost: per-batch reduction (fixed order, overwrites d_out) -------
__global__ void sk_cost_final_kernel(const float* __restrict__ rowsum, float* __restrict__ out) {
    int n = blockIdx.x;
    __shared__ float red[256];
    float s = 0.0f;
    for (int i = threadIdx.x; i < P; i += 256) s += rowsum[(size_t)n * P + i];
    red[threadIdx.x] = s;
    __syncthreads();
    for (int w = 128; w; w >>= 1) {
        if ((int)threadIdx.x < w) red[threadIdx.x] += red[threadIdx.x + w];
        __syncthreads();
    }
    if (threadIdx.x == 0) out[n] = red[0];
}

extern "C" void kernel_launch(void* const* d_in, const int* in_sizes, int n_in,
                              void* d_out, int out_size, void* d_ws, size_t ws_size,
                              hipStream_t stream) {
    const float* x = (const float*)d_in[0];
    const float* y = (const float*)d_in[1];
    float* out = (float*)d_out;

    // workspace layout (floats). C (134 MB) fits the 192 MB L2 -> iterations are L2-resident.
    float* ws = (float*)d_ws;
    const size_t NP = (size_t)NB * P;
    float* C      = ws;                        // NB*P*P
    float* xn     = C  + (size_t)NB * P * P;   // NP
    float* yn     = xn + NP;                   // NP
    float* u      = yn + NP;                   // NP
    float* v      = u  + NP;                   // NP
    float* du     = v  + NP;                   // NP
    float* pm     = du + NP;                   // RCHUNKS*NP
    float* ps     = pm + (size_t)RCHUNKS * NP; // RCHUNKS*NP
    float* rowsum = ps + (size_t)RCHUNKS * NP; // NP
    int*   done   = (int*)(rowsum + NP);       // 1

    // init u=v=0, done=0 (re-done every call -> deterministic across graph replays)
    sk_init_kernel<<<(NB * P + 255) / 256, 256, 0, stream>>>(u, v, done);

    // row norms of x and y
    sk_norms_kernel<<<(2 * NB * P) / 256, 256, 0, stream>>>(x, y, xn, yn);

    // cost matrix via f32 WMMA: 8 batches * 128x128 tiles, 8 waves/block
    sk_cost_wmma_kernel<<<(NB * 16384) / 8, 256, 0, stream>>>(x, y, xn, yn, C);

    // Sinkhorn iterations (device-side early exit mirrors the while_loop condition)
    for (int it = 0; it < MAX_ITER; ++it) {
        sk_update_u_kernel<<<(NB * P) / 8, 256, 0, stream>>>(C, u, v, du, done);
        sk_v_partial_kernel<<<16 * RCHUNKS, 256, 0, stream>>>(C, u, pm, ps, done);
        sk_v_combine_kernel<<<(NB * P) / 256, 256, 0, stream>>>(v, pm, ps, done);
        sk_check_kernel<<<1, 256, 0, stream>>>(du, done);
    }

    // cost = sum(exp(M) * C)
    sk_cost_partial_kernel<<<(NB * P) / 8, 256, 0, stream>>>(C, u, v, rowsum);
    sk_cost_final_kernel<<<NB, 256, 0, stream>>>(rowsum, out);
}